// GNNet_59596966199336
// MI455X (gfx1250) — compile-verified
//
#include <hip/hip_runtime.h>
#include <hip/hip_bf16.h>

#define DDIM    256
#define NNODES  10000
#define NEDGES  320000
#define NGRAPHS 64
#define NGROUP  32        // output columns per block (LDS-staged weight slice)
#define LDS_PITCH 260     // 260 % 64 == 4 -> conflict-free ds_load_b64 for B operand

typedef __attribute__((ext_vector_type(2))) float v2f;
typedef __attribute__((ext_vector_type(8))) float v8f;

// ---------------------------------------------------------------------------
// agg = h  (copy; folds the "+ h" of GIN eps=0 into the aggregation buffer)
// ---------------------------------------------------------------------------
__global__ void copy_f4(const float* __restrict__ src, float* __restrict__ dst, int n4) {
  int i = blockIdx.x * blockDim.x + threadIdx.x;
  if (i < n4) ((float4*)dst)[i] = ((const float4*)src)[i];
}

// ---------------------------------------------------------------------------
// agg[dst] += h[src] over all edges. 64 threads/edge, float4 gather + 4 fp32
// atomics. h (10.24 MB) is L2-resident (192 MB L2) so this is L2-bound.
// ---------------------------------------------------------------------------
__global__ void scatter_add_edges(const float* __restrict__ H,
                                  const int* __restrict__ srcIdx,
                                  const int* __restrict__ dstIdx,
                                  float* __restrict__ Agg) {
  int t = blockIdx.x * blockDim.x + threadIdx.x;
  int e = t >> 6;
  int p = (t & 63) << 2;
  if (e >= NEDGES) return;
  int s = srcIdx[e];
  int d = dstIdx[e];
  float4 v = *(const float4*)(H + (size_t)s * DDIM + p);
  float* o = Agg + (size_t)d * DDIM + p;
  atomicAdd(o + 0, v.x);
  atomicAdd(o + 1, v.y);
  atomicAdd(o + 2, v.z);
  atomicAdd(o + 3, v.w);
}

// ---------------------------------------------------------------------------
// Out = relu(A @ W + bias), A:[10000,256], W:[256,256].
// fp32 WMMA 16x16x4. One wave = 16x32 output tile (2 accumulators).
// W[:, nbase:nbase+32] staged transposed in LDS: ldsB[c*PITCH + k] = W[k][nbase+c]
// ---------------------------------------------------------------------------
__global__ __launch_bounds__(256) void gin_gemm_relu(
    const float* __restrict__ A, const float* __restrict__ W,
    const float* __restrict__ bias, float* __restrict__ Out) {
  __shared__ float ldsB[NGROUP * LDS_PITCH];
  const int tid   = threadIdx.x;
  const int nbase = blockIdx.x * NGROUP;

  // cooperative staging: 256 rows x 32 cols = 2048 float4 loads over 256 threads
  for (int i = tid; i < DDIM * (NGROUP / 4); i += 256) {
    int k  = i >> 3;          // row of W (K index)
    int c4 = (i & 7) << 2;    // column group within the 32-col slice
    float4 v = *(const float4*)(W + (size_t)k * DDIM + nbase + c4);
    ldsB[(c4 + 0) * LDS_PITCH + k] = v.x;
    ldsB[(c4 + 1) * LDS_PITCH + k] = v.y;
    ldsB[(c4 + 2) * LDS_PITCH + k] = v.z;
    ldsB[(c4 + 3) * LDS_PITCH + k] = v.w;
  }
  __syncthreads();

  const int lane  = tid & 31;
  const int wid   = tid >> 5;
  const int mtile = blockIdx.y * 8 + wid;
  if (mtile >= NNODES / 16) return;           // 625 tiles, grid covers 632 waves
  const int mbase = mtile * 16;
  const int l15   = lane & 15;
  const int khi   = (lane >> 4) << 1;         // 0 (lanes 0-15) or 2 (lanes 16-31)
  const float* arow = A + (size_t)(mbase + l15) * DDIM + khi;

  union V8 { v8f v; float f[8]; };
  V8 acc[2];
#pragma unroll
  for (int t = 0; t < 2; ++t) {
    // C/D layout: VGPR r holds M = r + 8*(lane>=16), N = lane&15 -> bias is N-only
    float bv = bias[nbase + t * 16 + l15];
#pragma unroll
    for (int r = 0; r < 8; ++r) acc[t].f[r] = bv;
  }

  const float* b0 = &ldsB[(0 * 16 + l15) * LDS_PITCH + khi];
  const float* b1 = &ldsB[(1 * 16 + l15) * LDS_PITCH + khi];

#pragma unroll 4
  for (int k0 = 0; k0 < DDIM; k0 += 4) {
    // A 16x4 f32 operand: lane<16 -> {A[m][k0],A[m][k0+1]}, lane>=16 -> {+2,+3}
    v2f a   = *(const v2f*)(arow + k0);
    v2f bb0 = *(const v2f*)(b0 + k0);   // B 4x16 operand from LDS (ds_load_b64)
    v2f bb1 = *(const v2f*)(b1 + k0);
    acc[0].v = __builtin_amdgcn_wmma_f32_16x16x4_f32(
        false, a, false, bb0, (short)0, acc[0].v, false, false);
    acc[1].v = __builtin_amdgcn_wmma_f32_16x16x4_f32(
        false, a, false, bb1, (short)0, acc[1].v, false, false);
  }

  const int rhi = (lane >> 4) << 3;  // 0 or 8
#pragma unroll
  for (int t = 0; t < 2; ++t) {
#pragma unroll
    for (int r = 0; r < 8; ++r) {
      float v = acc[t].f[r];
      v = v > 0.f ? v : 0.f;
      Out[(size_t)(mbase + rhi + r) * DDIM + nbase + t * 16 + l15] = v;
    }
  }
}

// ---------------------------------------------------------------------------
// Pooling: zero -> atomic accumulate (sum into G[:,0:256], max into G[:,256:512],
// int-reinterp atomicMax is exact for the non-negative post-ReLU values) -> mean.
// ---------------------------------------------------------------------------
__global__ void zero_pool(float* __restrict__ G, int* __restrict__ counts) {
  int i = blockIdx.x * blockDim.x + threadIdx.x;
  if (i < NGRAPHS * 512) G[i] = 0.f;
  if (i < NGRAPHS) counts[i] = 0;
}

__global__ void pool_accum(const float* __restrict__ H, const int* __restrict__ batch,
                           float* __restrict__ G, int* __restrict__ counts) {
  int t = blockIdx.x * blockDim.x + threadIdx.x;
  int node = t >> 6;
  int p = (t & 63) << 2;
  if (node >= NNODES) return;
  int b = batch[node];
  float4 v = *(const float4*)(H + (size_t)node * DDIM + p);
  float* gs = G + (size_t)b * 512 + p;
  atomicAdd(gs + 0, v.x);
  atomicAdd(gs + 1, v.y);
  atomicAdd(gs + 2, v.z);
  atomicAdd(gs + 3, v.w);
  int* gm = (int*)(G + (size_t)b * 512 + 256 + p);
  atomicMax(gm + 0, __float_as_int(v.x));
  atomicMax(gm + 1, __float_as_int(v.y));
  atomicMax(gm + 2, __float_as_int(v.z));
  atomicMax(gm + 3, __float_as_int(v.w));
  if (p == 0) atomicAdd(counts + b, 1);
}

__global__ void pool_final(float* __restrict__ G, const int* __restrict__ counts) {
  int i = blockIdx.x * blockDim.x + threadIdx.x;
  if (i >= NGRAPHS * 256) return;
  int g = i >> 8, c = i & 255;
  int cnt = counts[g];
  float inv = 1.f / (float)(cnt > 1 ? cnt : 1);
  G[(size_t)g * 512 + c] *= inv;
}

// ---------------------------------------------------------------------------
// MLP head: out[g,n] = act(sum_k in[g,k]*W[k,n] + b[n]); mode 1=relu, 2=sigmoid.
// Lanes stride over n -> W reads coalesced. Tiny (<= 19 MFLOP total).
// ---------------------------------------------------------------------------
__global__ void fc_kernel(const float* __restrict__ In, const float* __restrict__ W,
                          const float* __restrict__ B, float* __restrict__ Out,
                          int G, int K, int N, int mode) {
  int i = blockIdx.x * blockDim.x + threadIdx.x;
  if (i >= G * N) return;
  int g = i / N, n = i % N;
  const float* in = In + (size_t)g * K;
  float acc = B[n];
  for (int k = 0; k < K; ++k) acc = fmaf(in[k], W[(size_t)k * N + n], acc);
  if (mode == 1) acc = fmaxf(acc, 0.f);
  else           acc = 1.f / (1.f + __expf(-acc));
  Out[i] = acc;
}

// ---------------------------------------------------------------------------
extern "C" void kernel_launch(void* const* d_in, const int* in_sizes, int n_in,
                              void* d_out, int out_size, void* d_ws, size_t ws_size,
                              hipStream_t stream) {
  (void)in_sizes; (void)n_in; (void)out_size; (void)ws_size;

  const float* x     = (const float*)d_in[0];
  const int*   ei    = (const int*)d_in[1];   // [2, NEDGES]: src row, then dst row
  const int*   batch = (const int*)d_in[2];
  const float* Wl[5]; const float* Bl[5];
  for (int i = 0; i < 5; ++i) {
    Wl[i] = (const float*)d_in[3 + 2 * i];
    Bl[i] = (const float*)d_in[4 + 2 * i];
  }
  const float* wl1 = (const float*)d_in[13]; const float* bl1 = (const float*)d_in[14];
  const float* wl2 = (const float*)d_in[15]; const float* bl2 = (const float*)d_in[16];
  const float* wl3 = (const float*)d_in[17]; const float* bl3 = (const float*)d_in[18];
  float* out = (float*)d_out;

  float* agg = (float*)d_ws;                       // [10000,256]
  float* h   = agg + (size_t)NNODES * DDIM;        // [10000,256]
  float* g   = h   + (size_t)NNODES * DDIM;        // [64,512] mean|max
  float* g1  = g   + NGRAPHS * 512;                // [64,256]
  float* g2  = g1  + NGRAPHS * 256;                // [64,64]
  int*  counts = (int*)(g2 + NGRAPHS * 64);        // [64]

  const int* srcIdx = ei;
  const int* dstIdx = ei + NEDGES;

  const int n4 = NNODES * DDIM / 4;
  dim3 gemmGrid(DDIM / NGROUP, (NNODES / 16 + 7) / 8);   // (8, 79)

  const float* hin = x;
  for (int l = 0; l < 5; ++l) {
    copy_f4<<<(n4 + 255) / 256, 256, 0, stream>>>(hin, agg, n4);
    scatter_add_edges<<<(NEDGES * 64) / 256, 256, 0, stream>>>(hin, srcIdx, dstIdx, agg);
    gin_gemm_relu<<<gemmGrid, 256, 0, stream>>>(agg, Wl[l], Bl[l], h);
    hin = h;
  }

  zero_pool<<<(NGRAPHS * 512 + 255) / 256, 256, 0, stream>>>(g, counts);
  pool_accum<<<(NNODES * 64) / 256, 256, 0, stream>>>(h, batch, g, counts);
  pool_final<<<(NGRAPHS * 256 + 255) / 256, 256, 0, stream>>>(g, counts);

  fc_kernel<<<(NGRAPHS * 256 + 255) / 256, 256, 0, stream>>>(g,  wl1, bl1, g1, NGRAPHS, 512, 256, 1);
  fc_kernel<<<(NGRAPHS * 64  + 255) / 256, 256, 0, stream>>>(g1, wl2, bl2, g2, NGRAPHS, 256, 64, 1);
  fc_kernel<<<1, 64, 0, stream>>>(g2, wl3, bl3, out, NGRAPHS, 64, 1, 2);
}